// GraphConvBlock_61263413510654
// MI455X (gfx1250) — compile-verified
//
#include <hip/hip_runtime.h>

#define DDIM 128
#define NGRAPHS 8
#define EPS 1e-5f
#define PSTRIDE 144   // pair-row stride in float2 (=288 floats): 288%64==32 banks -> conflict-free

typedef float v2f __attribute__((ext_vector_type(2)));
typedef float v8f __attribute__((ext_vector_type(8)));

// ---------------------------------------------------------------- utilities
__device__ __forceinline__ void atomAddF(float* p, float v) {
    unsafeAtomicAdd(p, v);   // hardware global_atomic_add_f32
}

__global__ __launch_bounds__(256)
void zero_f_kernel(float* __restrict__ p, long n) {
    long i = (long)blockIdx.x * 256 + threadIdx.x;
    long stride = (long)gridDim.x * 256;
    for (; i < n; i += stride) p[i] = 0.0f;
}

__global__ __launch_bounds__(256)
void zero_f4_kernel(float4* __restrict__ p, long n4) {
    long i = (long)blockIdx.x * 256 + threadIdx.x;
    long stride = (long)gridDim.x * 256;
    float4 z = make_float4(0.f, 0.f, 0.f, 0.f);
    for (; i < n4; i += stride) p[i] = z;
}

// ---------------------------------------------------------------- degree
__global__ __launch_bounds__(256)
void deg_kernel(const int* __restrict__ dst, const float* __restrict__ ew,
                float* __restrict__ deg, int E) {
    int e = blockIdx.x * 256 + threadIdx.x;
    if (e < E) atomAddF(&deg[dst[e]], ew[e]);
}

__global__ __launch_bounds__(256)
void dinv_kernel(const float* __restrict__ deg, float* __restrict__ dinv, int N) {
    int n = blockIdx.x * 256 + threadIdx.x;
    if (n < N) dinv[n] = rsqrtf(deg[n] + 1.0f);
}

// ---------------------------------------------------------------- GEMM h = x @ W  (fp32 WMMA)
// block = 256 thr = 8 waves; wave w -> rows rb..rb+15, cols 0..127 (8 wmma acc).
// W staged in LDS K-pair-interleaved: Wl[p][c] = {W[2p][c], W[2p+1][c]} so each
// B fragment is ONE aligned ds_load_b64 straight into a wmma-ready pair.
__global__ __launch_bounds__(256)
void gemm_xw_kernel(const float* __restrict__ x, const float* __restrict__ W,
                    float* __restrict__ h, int nrows) {
    __shared__ v2f Wl[(DDIM / 2) * PSTRIDE];   // 64 * 144 * 8B = 72 KB
    const int tid = threadIdx.x;

    // stage: thread i -> pair p = i>>7, col c = i&127 (coalesced 512B row reads)
    for (int i = tid; i < (DDIM / 2) * DDIM; i += 256) {
        int p = i >> 7;
        int c = i & 127;
        v2f w2;
        w2.x = W[(2 * p    ) * DDIM + c];
        w2.y = W[(2 * p + 1) * DDIM + c];
        Wl[p * PSTRIDE + c] = w2;
    }
    __syncthreads();

    const int wave = tid >> 5;
    const int lane = tid & 31;
    const int half = lane >> 4;     // 0: K pair {k0,k0+1}, 1: K pair {k0+2,k0+3}
    const int lr   = lane & 15;
    const int rb   = blockIdx.x * 128 + wave * 16;
    if (rb >= nrows) return;        // uniform per-wave: EXEC stays all-ones

    v8f acc[8] = {};

    const float* xrow = x + (long)(rb + lr) * DDIM;        // A: lane holds row rb+lr
    const v2f* __restrict__ Bbase = &Wl[half * PSTRIDE + lr]; // per-lane LDS base

#pragma unroll 8
    for (int k0 = 0; k0 < DDIM; k0 += 4) {
        const int kb = k0 + 2 * half;
        v2f a = *reinterpret_cast<const v2f*>(xrow + kb);  // K = kb, kb+1
        const v2f* Brow = Bbase + (k0 >> 1) * PSTRIDE;     // pair row k0/2 (+half)
#pragma unroll
        for (int t = 0; t < 8; ++t) {
            v2f bf = Brow[t * 16];                          // one ds_load_b64
            acc[t] = __builtin_amdgcn_wmma_f32_16x16x4_f32(
                false, a, false, bf, (short)0, acc[t], false, false);
        }
    }

    // C/D layout: lanes 0-15 -> rows rb+0..7 (vgpr r), lanes 16-31 -> rows rb+8..15
#pragma unroll
    for (int t = 0; t < 8; ++t) {
        const int cb = t * 16;
#pragma unroll
        for (int r = 0; r < 8; ++r) {
            h[(long)(rb + half * 8 + r) * DDIM + cb + lr] = acc[t][r];
        }
    }
}

// ---------------------------------------------------------------- edge scatter (1 wave / edge)
__global__ __launch_bounds__(256)
void scatter_kernel(const float* __restrict__ h, const int* __restrict__ src,
                    const int* __restrict__ dst, const float* __restrict__ ew,
                    const float* __restrict__ dinv, float* __restrict__ agg, int E) {
    int e    = (int)(((long)blockIdx.x * 256 + threadIdx.x) >> 5);
    int lane = threadIdx.x & 31;
    if (e >= E) return;
    int s = src[e], d = dst[e];
    float en = dinv[s] * ew[e] * dinv[d];
    float4 v = reinterpret_cast<const float4*>(h)[(long)s * 32 + lane];
    float* o = agg + (long)d * DDIM + lane * 4;
    atomAddF(o + 0, v.x * en);
    atomAddF(o + 1, v.y * en);
    atomAddF(o + 2, v.z * en);
    atomAddF(o + 3, v.w * en);
}

// ---------------------------------------------------------------- per-graph counts (batch sorted)
__global__ void cnt_kernel(const int* __restrict__ batch, float* __restrict__ cnt, int N) {
    int g = threadIdx.x;
    if (g >= NGRAPHS) return;
    auto lb = [&](int key) {
        int lo = 0, hi = N;
        while (lo < hi) { int mid = (lo + hi) >> 1; if (batch[mid] < key) lo = mid + 1; else hi = mid; }
        return lo;
    };
    int c = lb(g + 1) - lb(g);
    cnt[g] = fmaxf((float)c, 1.0f);
}

// ---------------------------------------------------------------- finalize + segmented graph sums
// block = 128 thr (thread = feature dim), 64 nodes per block, batch sorted => few flushes
#define NPB 64
__global__ __launch_bounds__(128)
void finalize_kernel(const float* __restrict__ h, const float* __restrict__ dinv,
                     const float* __restrict__ bvec, const int* __restrict__ batch,
                     float* __restrict__ agg, float* __restrict__ gsum,
                     float* __restrict__ gsq, int N) {
    const int dcol = threadIdx.x;
    const int base = blockIdx.x * NPB;
    const float bb = bvec[dcol];
    float s = 0.f, q = 0.f;
    int cur = -1;
    for (int i = 0; i < NPB; ++i) {
        int n = base + i;
        if (n >= N) break;
        int g = batch[n];
        float di = dinv[n];
        long idx = (long)n * DDIM + dcol;
        float a = agg[idx] + h[idx] * (di * di) + bb;   // + self-loop term + bias
        agg[idx] = a;
        if (g != cur) {
            if (cur >= 0) {
                atomAddF(&gsum[cur * DDIM + dcol], s);
                atomAddF(&gsq [cur * DDIM + dcol], q);
            }
            cur = g; s = a; q = a * a;
        } else { s += a; q += a * a; }
    }
    if (cur >= 0) {
        atomAddF(&gsum[cur * DDIM + dcol], s);
        atomAddF(&gsq [cur * DDIM + dcol], q);
    }
}

// ---------------------------------------------------------------- GraphNorm + exact GELU
// var = E[a^2] - 2*s*m^2 + s^2*m^2  (single-pass moments)
__global__ __launch_bounds__(256)
void norm_gelu_kernel(const int* __restrict__ batch, const float* __restrict__ gsum,
                      const float* __restrict__ gsq, const float* __restrict__ cnt,
                      const float* __restrict__ gw, const float* __restrict__ gb,
                      const float* __restrict__ gms, float* __restrict__ agg, int N) {
    long t = (long)blockIdx.x * 256 + threadIdx.x;   // one float4 per thread
    long total = (long)N * (DDIM / 4);
    if (t >= total) return;
    int n  = (int)(t >> 5);
    int c4 = ((int)t & 31) << 2;
    int g  = batch[n];
    float ic = 1.0f / cnt[g];
    float4 a = reinterpret_cast<float4*>(agg)[t];
    float r[4] = {a.x, a.y, a.z, a.w};
#pragma unroll
    for (int j = 0; j < 4; ++j) {
        int d = c4 + j;
        float m  = gsum[g * DDIM + d] * ic;
        float e2 = gsq [g * DDIM + d] * ic;
        float sc = gms[d];
        float c  = r[j] - sc * m;
        float var = e2 - 2.0f * sc * m * m + sc * sc * m * m;
        float o = gw[d] * c * rsqrtf(var + EPS) + gb[d];
        r[j] = 0.5f * o * (1.0f + erff(o * 0.70710678118654752f));   // exact GELU
    }
    reinterpret_cast<float4*>(agg)[t] = make_float4(r[0], r[1], r[2], r[3]);
}

// ---------------------------------------------------------------- launch
extern "C" void kernel_launch(void* const* d_in, const int* in_sizes, int n_in,
                              void* d_out, int out_size, void* d_ws, size_t ws_size,
                              hipStream_t stream) {
    const float* x     = (const float*)d_in[0];
    const int*   eidx  = (const int*)  d_in[1];
    const float* ew    = (const float*)d_in[2];
    const int*   batch = (const int*)  d_in[3];
    const float* W     = (const float*)d_in[4];
    const float* b     = (const float*)d_in[5];
    const float* gw    = (const float*)d_in[6];
    const float* gb    = (const float*)d_in[7];
    const float* gms   = (const float*)d_in[8];
    float* out = (float*)d_out;

    const int N = in_sizes[0] / DDIM;
    const int E = in_sizes[2];
    const int* src = eidx;
    const int* dst = eidx + E;

    // workspace layout (floats): deg | gsum | gsq | cnt | dinv | h
    float* w = (float*)d_ws;
    const long off_deg  = 0;
    const long off_gsum = off_deg + N;
    const long off_gsq  = off_gsum + NGRAPHS * DDIM;
    const long off_cnt  = off_gsq + NGRAPHS * DDIM;
    const long off_dinv = off_cnt + NGRAPHS;
    long off_h = off_dinv + N;
    off_h = (off_h + 3) & ~3L;               // float4-align h
    float* deg  = w + off_deg;
    float* gsum = w + off_gsum;
    float* gsq  = w + off_gsq;
    float* cnt  = w + off_cnt;
    float* dinv = w + off_dinv;
    float* h    = w + off_h;

    // 1) zero: [deg..cnt] contiguous, and agg (= d_out)
    long nz = off_dinv;                       // deg + gsum + gsq + cnt
    zero_f_kernel <<<(int)((nz + 255) / 256), 256, 0, stream>>>(w, nz);
    long n4 = (long)N * (DDIM / 4);
    zero_f4_kernel<<<(int)((n4 + 255) / 256), 256, 0, stream>>>((float4*)out, n4);

    // 2) degree + dinv
    deg_kernel <<<(E + 255) / 256, 256, 0, stream>>>(dst, ew, deg, E);
    dinv_kernel<<<(N + 255) / 256, 256, 0, stream>>>(deg, dinv, N);

    // 3) h = x @ W via v_wmma_f32_16x16x4_f32
    gemm_xw_kernel<<<(N + 127) / 128, 256, 0, stream>>>(x, W, h, N);

    // 4) edge scatter: agg[dst] += h[src] * enorm  (1 wave / edge)
    scatter_kernel<<<(int)(((long)E * 32 + 255) / 256), 256, 0, stream>>>(h, src, dst, ew, dinv, out, E);

    // 5) per-graph counts via binary search on sorted batch
    cnt_kernel<<<1, 32, 0, stream>>>(batch, cnt, N);

    // 6) add self-loop + bias, accumulate per-graph moments
    finalize_kernel<<<(N + NPB - 1) / NPB, 128, 0, stream>>>(h, dinv, b, batch, out, gsum, gsq, N);

    // 7) GraphNorm + exact GELU, in place on d_out
    norm_gelu_kernel<<<(int)((n4 + 255) / 256), 256, 0, stream>>>(batch, gsum, gsq, cnt, gw, gb, gms, out, N);
}